// GCNEncoder_10342281249035
// MI455X (gfx1250) — compile-verified
//
#include <hip/hip_runtime.h>

typedef __attribute__((ext_vector_type(2))) float v2f;
typedef __attribute__((ext_vector_type(8))) float v8f;

#define FHID 128  // hidden width; all aggregation buffers are [N,128]

// ---------------- degree / norm precompute ----------------

__global__ void k_deg_init(float* __restrict__ deg, int n) {
  int i = blockIdx.x * blockDim.x + threadIdx.x;
  if (i < n) deg[i] = 1.0f;  // self-loop contributes 1 to every node's degree
}

__global__ void k_deg_accum(const int* __restrict__ dst, float* __restrict__ deg, int e) {
  int i = blockIdx.x * blockDim.x + threadIdx.x;
  if (i < e) atomicAdd(&deg[dst[i]], 1.0f);
}

__global__ void k_dinv(float* __restrict__ deg, int n) {
  int i = blockIdx.x * blockDim.x + threadIdx.x;
  if (i < n) deg[i] = rsqrtf(deg[i]);  // deg >= 1 always
}

__global__ void k_norm(const int* __restrict__ src, const int* __restrict__ dst,
                       const float* __restrict__ dinv, float* __restrict__ norm, int e) {
  int i = blockIdx.x * blockDim.x + threadIdx.x;
  if (i < e) norm[i] = dinv[src[i]] * dinv[dst[i]];
}

// ---------------- dense GEMM via V_WMMA_F32_16X16X4_F32 ----------------
// T[tm*16.., colOff + tn*16..] = X[tm*16.., 0:128] @ W[0:128, tn*16..]
// One wave computes one 16x16 f32 tile; K=128 -> 32 WMMA ops.
// A (16x4 f32): lane m=lane&15 holds rows; VGPR0/1 = K{0,1} for lanes 0-15,
// K{2,3} for lanes 16-31.  C/D: VGPR r -> row r (lanes 0-15) / row 8+r (16-31).
__global__ void k_gemm_wmma(const float* __restrict__ X, const float* __restrict__ W,
                            float* __restrict__ T, int tilesM, int Fout, int ldT, int colOff) {
  const int K = FHID;
  int gwave = (blockIdx.x * blockDim.x + threadIdx.x) >> 5;
  int lane  = threadIdx.x & 31;
  int tilesN = Fout >> 4;
  int tm = gwave / tilesN;
  int tn = gwave - tm * tilesN;
  if (tm >= tilesM) return;  // wave-uniform exit: EXEC stays all-ones for WMMA

  int m  = lane & 15;
  int kh = lane >> 4;  // which K-pair this half-wave holds

  const float* Xr = X + (size_t)(tm * 16 + m) * K + 2 * kh;          // A: row m, K offset
  const float* Wc = W + (size_t)(2 * kh) * Fout + tn * 16 + m;       // B: col m, K offset

  v8f acc = {};
  for (int k0 = 0; k0 < K; k0 += 4) {
    v2f a = *(const v2f*)(Xr + k0);          // X[row, k0+2kh], X[row, k0+2kh+1]
    v2f b;
    b.x = Wc[(size_t)k0 * Fout];             // W[k0+2kh,   col]
    b.y = Wc[(size_t)(k0 + 1) * Fout];       // W[k0+2kh+1, col]
    acc = __builtin_amdgcn_wmma_f32_16x16x4_f32(
        /*neg_a=*/false, a, /*neg_b=*/false, b,
        /*c_mod=*/(short)0, acc, /*reuse_a=*/false, /*reuse_b=*/false);
  }

  float* Trow = T + (size_t)(tm * 16 + kh * 8) * ldT + colOff + tn * 16 + m;
#pragma unroll
  for (int r = 0; r < 8; ++r) Trow[(size_t)r * ldT] = acc[r];
}

// ---------------- aggregation (self-loop init, edge scatter, finalize) ----------------

__global__ void k_self_init(const float* __restrict__ T, const float* __restrict__ dinv,
                            float* __restrict__ out, long total) {
  long idx = (long)blockIdx.x * blockDim.x + threadIdx.x;
  if (idx >= total) return;
  int i = (int)(idx >> 7);
  float di = dinv[i];
  out[idx] = di * di * T[idx];  // self-loop term; also overwrites ws poison
}

__global__ void k_scatter(const int* __restrict__ src, const int* __restrict__ dst,
                          const float* __restrict__ norm, const float* __restrict__ T,
                          float* __restrict__ out, long total) {
  long idx = (long)blockIdx.x * blockDim.x + threadIdx.x;
  if (idx >= total) return;
  int e = (int)(idx >> 7);
  int f = (int)(idx & 127);
  int s = src[e], d = dst[e];
  atomicAdd(&out[(size_t)d * FHID + f], norm[e] * T[(size_t)s * FHID + f]);
}

__global__ void k_bias_relu(float* __restrict__ h, const float* __restrict__ b, long total) {
  long idx = (long)blockIdx.x * blockDim.x + threadIdx.x;
  if (idx >= total) return;
  h[idx] = fmaxf(h[idx] + b[idx & 127], 0.0f);
}

// agg is [N,128] with cols 0..63 = mu-pre-bias, 64..127 = logvar-pre-bias.
// d_out = mu [N,64] flat, then logvar [N,64] flat.
__global__ void k_finalize_mulv(const float* __restrict__ agg, const float* __restrict__ bmu,
                                const float* __restrict__ blv, float* __restrict__ out, int n) {
  long idx = (long)blockIdx.x * blockDim.x + threadIdx.x;
  long total = (long)n * FHID;
  if (idx >= total) return;
  int i = (int)(idx >> 7);
  int f = (int)(idx & 127);
  float v = agg[idx];
  if (f < 64) out[(size_t)i * 64 + f] = v + bmu[f];
  else        out[(size_t)n * 64 + (size_t)i * 64 + (f - 64)] = v + blv[f - 64];
}

// ---------------- host-side orchestration ----------------

extern "C" void kernel_launch(void* const* d_in, const int* in_sizes, int n_in,
                              void* d_out, int out_size, void* d_ws, size_t ws_size,
                              hipStream_t stream) {
  const float* x   = (const float*)d_in[0];
  const int*   ei  = (const int*)d_in[1];
  const float* W1  = (const float*)d_in[2];
  const float* b1  = (const float*)d_in[3];
  const float* W2  = (const float*)d_in[4];
  const float* b2  = (const float*)d_in[5];
  const float* Wmu = (const float*)d_in[6];
  const float* bmu = (const float*)d_in[7];
  const float* Wlv = (const float*)d_in[8];
  const float* blv = (const float*)d_in[9];

  const int n = in_sizes[0] / FHID;   // 50000
  const int e = in_sizes[1] / 2;      // 800000
  const int* src = ei;                // edge_index row 0
  const int* dst = ei + e;            // edge_index row 1

  // workspace carve-up (256B aligned)
  char* ws = (char*)d_ws;
  size_t off = 0;
  auto alloc = [&](size_t bytes) -> void* {
    void* p = ws + off;
    off += (bytes + 255) & ~(size_t)255;
    return p;
  };
  float* dinv = (float*)alloc((size_t)n * sizeof(float));
  float* norm = (float*)alloc((size_t)e * sizeof(float));
  float* bufT = (float*)alloc((size_t)n * FHID * sizeof(float));  // GEMM output
  float* bufA = (float*)alloc((size_t)n * FHID * sizeof(float));  // agg ping
  float* bufB = (float*)alloc((size_t)n * FHID * sizeof(float));  // agg pong

  const int BS = 256;
  auto nblk = [](long tot, int bs) { return (unsigned)((tot + bs - 1) / bs); };
  const long totN = (long)n * FHID;   // 6.4M
  const long totE = (long)e * FHID;   // 102.4M

  auto gemm = [&](const float* hin, const float* W, int Fout, float* T, int colOff) {
    int tilesM = n / 16;                         // 3125 (exact)
    long waves = (long)tilesM * (Fout >> 4);
    unsigned blocks = (unsigned)((waves + 7) / 8);  // 8 waves (256 threads) / block
    k_gemm_wmma<<<blocks, 256, 0, stream>>>(hin, W, T, tilesM, Fout, FHID, colOff);
  };

  // ---- normalization coefficients ----
  k_deg_init <<<nblk(n, BS), BS, 0, stream>>>(dinv, n);
  k_deg_accum<<<nblk(e, BS), BS, 0, stream>>>(dst, dinv, e);
  k_dinv     <<<nblk(n, BS), BS, 0, stream>>>(dinv, n);
  k_norm     <<<nblk(e, BS), BS, 0, stream>>>(src, dst, dinv, norm, e);

  // ---- layer 1: h1 = relu(Agg(x @ W1) + b1) -> bufA ----
  gemm(x, W1, FHID, bufT, 0);
  k_self_init<<<nblk(totN, BS), BS, 0, stream>>>(bufT, dinv, bufA, totN);
  k_scatter  <<<nblk(totE, BS), BS, 0, stream>>>(src, dst, norm, bufT, bufA, totE);
  k_bias_relu<<<nblk(totN, BS), BS, 0, stream>>>(bufA, b1, totN);

  // ---- layer 2: h2 = relu(Agg(h1 @ W2) + b2) -> bufB ----
  gemm(bufA, W2, FHID, bufT, 0);
  k_self_init<<<nblk(totN, BS), BS, 0, stream>>>(bufT, dinv, bufB, totN);
  k_scatter  <<<nblk(totE, BS), BS, 0, stream>>>(src, dst, norm, bufT, bufB, totE);
  k_bias_relu<<<nblk(totN, BS), BS, 0, stream>>>(bufB, b2, totN);

  // ---- heads: pack [h2@Wmu | h2@Wlv] into one [N,128] buffer, one fused aggregation ----
  gemm(bufB, Wmu, 64, bufT, 0);
  gemm(bufB, Wlv, 64, bufT, 64);
  k_self_init<<<nblk(totN, BS), BS, 0, stream>>>(bufT, dinv, bufA, totN);
  k_scatter  <<<nblk(totE, BS), BS, 0, stream>>>(src, dst, norm, bufT, bufA, totE);
  k_finalize_mulv<<<nblk(totN, BS), BS, 0, stream>>>(bufA, bmu, blv, (float*)d_out, n);
}